// Model_34514357190974
// MI455X (gfx1250) — compile-verified
//
#include <hip/hip_runtime.h>
#include <math.h>

// ---------------- model constants ----------------
#define Bdim   8
#define Ldim   512
#define Ndim   1024
#define Ddim   128
#define Pdim   16
#define Tdim   32      // L / P
#define SCdim  4
#define Sdim   8       // T / SC
#define Rdim   16
#define Kdim   8
#define PREDdim 96

typedef __attribute__((ext_vector_type(2))) float v2f;
typedef __attribute__((ext_vector_type(8))) float v8f;

// ---------------------------------------------------------------
// 1) xbar[b,t,n] = mean_{p<16} x_enc[b, t*16+p, n]
// ---------------------------------------------------------------
__global__ void pool_kernel(const float* __restrict__ x, float* __restrict__ xbar) {
  int bt = blockIdx.x;                 // b*T + t
  int b = bt / Tdim, t = bt % Tdim;
  const float* base = x + ((size_t)b * Ldim + (size_t)t * Pdim) * Ndim;
  for (int n = threadIdx.x; n < Ndim; n += blockDim.x) {
    float s = 0.f;
#pragma unroll
    for (int p = 0; p < Pdim; ++p) s += base[(size_t)p * Ndim + n];
    xbar[(size_t)bt * Ndim + n] = s * (1.0f / Pdim);
  }
}

// ---------------------------------------------------------------
// 2) zbar[b,s,n] = mean_{c<4} xbar[b, s*4+c, n]
// ---------------------------------------------------------------
__global__ void zseg_kernel(const float* __restrict__ xbar, float* __restrict__ zbar) {
  int bs = blockIdx.x;                 // b*S + s
  int b = bs / Sdim, s = bs % Sdim;
  const float* base = xbar + ((size_t)b * Tdim + (size_t)s * SCdim) * Ndim;
  for (int n = threadIdx.x; n < Ndim; n += blockDim.x) {
    float v = 0.f;
#pragma unroll
    for (int c = 0; c < SCdim; ++c) v += base[(size_t)c * Ndim + n];
    zbar[(size_t)bs * Ndim + n] = v * 0.25f;
  }
}

// ---------------------------------------------------------------
// 3) fold graph-learner weights into 4 scalars: scores = a_n*a_m*al + a_n*be + a_m*ga + de
//    scal = {alpha, beta, gamma, delta, g=sigmoid(gate)}
// ---------------------------------------------------------------
__global__ void scal_kernel(const float* __restrict__ enc_w, const float* __restrict__ enc_b,
                            const float* __restrict__ gl_w1, const float* __restrict__ gl_b1,
                            const float* __restrict__ gl_w2, const float* __restrict__ gl_b2,
                            const float* __restrict__ gate, float* __restrict__ scal) {
  __shared__ float w1[Rdim], c1[Rdim], w2[Rdim], c2[Rdim];
  int r = threadIdx.x;
  if (r < Rdim) {
    float a = 0.f, bb = 0.f, cc = 0.f, dd = 0.f;
    for (int d = 0; d < Ddim; ++d) {
      float g1 = gl_w1[d * Rdim + r], g2 = gl_w2[d * Rdim + r];
      a  += enc_w[d] * g1;  bb += enc_b[d] * g1;
      cc += enc_w[d] * g2;  dd += enc_b[d] * g2;
    }
    w1[r] = a; c1[r] = bb + gl_b1[r]; w2[r] = cc; c2[r] = dd + gl_b2[r];
  }
  __syncthreads();
  if (threadIdx.x == 0) {
    float al = 0.f, be = 0.f, ga = 0.f, de = 0.f;
    for (int i = 0; i < Rdim; ++i) {
      al += w1[i] * w2[i]; be += w1[i] * c2[i];
      ga += c1[i] * w2[i]; de += c1[i] * c2[i];
    }
    scal[0] = al; scal[1] = be; scal[2] = ga; scal[3] = de;
    scal[4] = 1.0f / (1.0f + expf(-gate[0]));
  }
}

// ---------------------------------------------------------------
// 4) Wh[t,p] = sum_d enc_w[d] * head_w[t*D+d, p]   (folded head weights)
// ---------------------------------------------------------------
__global__ void whprep_kernel(const float* __restrict__ enc_w,
                              const float* __restrict__ head_w,
                              float* __restrict__ Wh) {
  int idx = blockIdx.x * blockDim.x + threadIdx.x;   // t*PRED + p
  if (idx >= Tdim * PREDdim) return;
  int t = idx / PREDdim, p = idx % PREDdim;
  const float* col = head_w + (size_t)t * Ddim * PREDdim + p;
  float s = 0.f;
  for (int d = 0; d < Ddim; ++d) s += enc_w[d] * col[(size_t)d * PREDdim];
  Wh[idx] = s;
}

// ---------------------------------------------------------------
// 5) Cp[p] = (1+g) * sum_{t,d} enc_b[d]*head_w[t*D+d, p] + head_b[p]
// ---------------------------------------------------------------
__global__ void cprep_kernel(const float* __restrict__ enc_b,
                             const float* __restrict__ head_w,
                             const float* __restrict__ head_b,
                             const float* __restrict__ gate,
                             float* __restrict__ Cp) {
  int p = threadIdx.x;
  if (p >= PREDdim) return;
  float g = 1.0f / (1.0f + expf(-gate[0]));
  float s = 0.f;
  for (int i = 0; i < Tdim * Ddim; ++i)
    s += enc_b[i & (Ddim - 1)] * head_w[(size_t)i * PREDdim + p];
  Cp[p] = (1.0f + g) * s + head_b[p];
}

// ---------------------------------------------------------------
// 6) per (b,s): top-8 largest and 8 smallest zbar values+indices
//    (ties -> lower index, matching jax.lax.top_k)
//    topv/topi layout: [b*S+s][0..7]=top desc, [8..15]=bottom asc
// ---------------------------------------------------------------
__global__ void topk_kernel(const float* __restrict__ zbar,
                            float* __restrict__ topv, int* __restrict__ topi) {
  int bs = blockIdx.x;
  __shared__ float a[Ndim];
  __shared__ float rv[256];
  __shared__ int   ri[256];
  const float* row = zbar + (size_t)bs * Ndim;
  int tid = threadIdx.x;
  for (int n = tid; n < Ndim; n += 256) a[n] = row[n];
  __syncthreads();
  // --- top 8 (argmax) ---
  for (int j = 0; j < Kdim; ++j) {
    float bv = -__builtin_inff(); int bi = Ndim;
    for (int n = tid; n < Ndim; n += 256) {
      float v = a[n];
      if (v > bv || (v == bv && n < bi)) { bv = v; bi = n; }
    }
    rv[tid] = bv; ri[tid] = bi; __syncthreads();
    for (int off = 128; off > 0; off >>= 1) {
      if (tid < off) {
        float v2 = rv[tid + off]; int i2 = ri[tid + off];
        if (v2 > rv[tid] || (v2 == rv[tid] && i2 < ri[tid])) { rv[tid] = v2; ri[tid] = i2; }
      }
      __syncthreads();
    }
    if (tid == 0) {
      topv[bs * 16 + j] = rv[0]; topi[bs * 16 + j] = ri[0];
      a[ri[0]] = -__builtin_inff();
    }
    __syncthreads();
  }
  // restore the 8 removed values
  if (tid < Kdim) a[topi[bs * 16 + tid]] = topv[bs * 16 + tid];
  __syncthreads();
  // --- bottom 8 (argmin) ---
  for (int j = 0; j < Kdim; ++j) {
    float bv = __builtin_inff(); int bi = Ndim;
    for (int n = tid; n < Ndim; n += 256) {
      float v = a[n];
      if (v < bv || (v == bv && n < bi)) { bv = v; bi = n; }
    }
    rv[tid] = bv; ri[tid] = bi; __syncthreads();
    for (int off = 128; off > 0; off >>= 1) {
      if (tid < off) {
        float v2 = rv[tid + off]; int i2 = ri[tid + off];
        if (v2 < rv[tid] || (v2 == rv[tid] && i2 < ri[tid])) { rv[tid] = v2; ri[tid] = i2; }
      }
      __syncthreads();
    }
    if (tid == 0) {
      topv[bs * 16 + 8 + j] = rv[0]; topi[bs * 16 + 8 + j] = ri[0];
      a[ri[0]] = __builtin_inff();
    }
    __syncthreads();
  }
}

// ---------------------------------------------------------------
// 7) gated graph mix:  U[(b*N+n)*T + t] = xbar + g * softmax8(c*a_mk) . xbar[m_k]
//    c = alpha*a_n + gamma selects top (c>=0) vs bottom (c<0) list.
// ---------------------------------------------------------------
__global__ void mix_kernel(const float* __restrict__ xbar, const float* __restrict__ zbar,
                           const float* __restrict__ topv, const int* __restrict__ topi,
                           const float* __restrict__ scal, float* __restrict__ U) {
  int bs = blockIdx.x; int b = bs / Sdim, s = bs % Sdim;
  __shared__ float lv[16];
  __shared__ int   li[16];
  __shared__ float xg[2][SCdim][Kdim];   // gathered xbar values, shared by all rows
  int tid = threadIdx.x;
  if (tid < 16) { lv[tid] = topv[bs * 16 + tid]; li[tid] = topi[bs * 16 + tid]; }
  __syncthreads();
  if (tid < 64) {
    int l = tid >> 5, c = (tid >> 3) & 3, k = tid & 7;
    int m = li[l * 8 + k];
    xg[l][c][k] = xbar[((size_t)b * Tdim + (size_t)(s * SCdim + c)) * Ndim + m];
  }
  __syncthreads();
  float alpha = scal[0], gamma = scal[2], g = scal[4];
  for (int n = tid; n < Ndim; n += blockDim.x) {
    float an = zbar[(size_t)bs * Ndim + n];
    float cf = alpha * an + gamma;
    int sel = (cf >= 0.f) ? 0 : 1;
    float w[Kdim]; float mx = -__builtin_inff();
#pragma unroll
    for (int k = 0; k < Kdim; ++k) { float t = cf * lv[sel * 8 + k]; w[k] = t; mx = fmaxf(mx, t); }
    float sum = 0.f;
#pragma unroll
    for (int k = 0; k < Kdim; ++k) { w[k] = expf(w[k] - mx); sum += w[k]; }
    float ginv = g / sum;
    float* urow = U + ((size_t)b * Ndim + n) * Tdim + (size_t)s * SCdim;
#pragma unroll
    for (int c = 0; c < SCdim; ++c) {
      float mixv = 0.f;
#pragma unroll
      for (int k = 0; k < Kdim; ++k) mixv += w[k] * xg[sel][c][k];
      float xb = xbar[((size_t)b * Tdim + (size_t)(s * SCdim + c)) * Ndim + n];
      urow[c] = xb + ginv * mixv;
    }
  }
}

// ---------------------------------------------------------------
// 8) head GEMM with f32 WMMA:  y[b,p,n] = U[b*N+n, :] @ Wh[:, p] + Cp[p]
//    M=8192, N=96, K=32; one wave per 16x16 tile; 8x chained 16x16x4 f32 WMMA.
// ---------------------------------------------------------------
__global__ void head_kernel(const float* __restrict__ U, const float* __restrict__ Wh,
                            const float* __restrict__ Cp, float* __restrict__ out) {
  const int NT = PREDdim / 16;                               // 6 tiles along PRED
  int wave = (blockIdx.x * blockDim.x + threadIdx.x) >> 5;   // global wave id = tile id
  int lane = threadIdx.x & 31;
  int tm = wave / NT, tn = wave % NT;
  if (tm >= (Bdim * Ndim) / 16) return;                      // never taken (exact grid)
  int rlo = lane & 15, hi = lane >> 4;
  int p = tn * 16 + rlo;
  float cp = Cp[p];
#if __has_builtin(__builtin_amdgcn_wmma_f32_16x16x4_f32)
  v8f acc = {};
  const float* Urow = U + (size_t)(tm * 16 + rlo) * Tdim;    // A: M = lane&15 (both halves)
#pragma unroll
  for (int ks = 0; ks < Tdim / 4; ++ks) {                    // K = 32 in 8 slices of 4
    int k0 = ks * 4 + hi * 2;                                // VGPR0->K{0,2}, VGPR1->K{1,3}
    v2f a;  a.x = Urow[k0];                 a.y = Urow[k0 + 1];
    v2f bb; bb.x = Wh[k0 * PREDdim + p];    bb.y = Wh[(k0 + 1) * PREDdim + p];
    acc = __builtin_amdgcn_wmma_f32_16x16x4_f32(
        /*neg_a=*/false, a, /*neg_b=*/false, bb,
        /*c_mod=*/(short)0, acc, /*reuse_a=*/false, /*reuse_b=*/false);
  }
#pragma unroll
  for (int v = 0; v < 8; ++v) {                              // D: VGPR v -> M = v + 8*hi
    int row = tm * 16 + v + 8 * hi;
    int b = row >> 10, n = row & (Ndim - 1);
    out[((size_t)b * PREDdim + p) * Ndim + n] = acc[v] + cp; // fused transpose to [B,PRED,N]
  }
#else
  for (int v = 0; v < 8; ++v) {
    int row = tm * 16 + v + 8 * hi;
    const float* Ur = U + (size_t)row * Tdim;
    float sacc = 0.f;
    for (int k = 0; k < Tdim; ++k) sacc += Ur[k] * Wh[k * PREDdim + p];
    int b = row >> 10, n = row & (Ndim - 1);
    out[((size_t)b * PREDdim + p) * Ndim + n] = sacc + cp;
  }
#endif
}

// ---------------------------------------------------------------
extern "C" void kernel_launch(void* const* d_in, const int* in_sizes, int n_in,
                              void* d_out, int out_size, void* d_ws, size_t ws_size,
                              hipStream_t stream) {
  const float* x_enc  = (const float*)d_in[0];
  const float* enc_w  = (const float*)d_in[4];
  const float* enc_b  = (const float*)d_in[5];
  const float* gl_w1  = (const float*)d_in[6];
  const float* gl_b1  = (const float*)d_in[7];
  const float* gl_w2  = (const float*)d_in[8];
  const float* gl_b2  = (const float*)d_in[9];
  const float* gate   = (const float*)d_in[10];
  const float* head_w = (const float*)d_in[11];
  const float* head_b = (const float*)d_in[12];
  float* out = (float*)d_out;

  float* ws   = (float*)d_ws;
  float* xbar = ws;                                  // B*T*N = 262144
  float* zbar = xbar + (size_t)Bdim * Tdim * Ndim;   // B*S*N =  65536
  float* U    = zbar + (size_t)Bdim * Sdim * Ndim;   // B*N*T = 262144
  float* Wh   = U    + (size_t)Bdim * Ndim * Tdim;   // T*PRED =  3072
  float* Cp   = Wh   + Tdim * PREDdim;               // 128
  float* scal = Cp   + 128;                          // 16
  float* topv = scal + 16;                           // B*S*16 = 1024
  int*   topi = (int*)(topv + Bdim * Sdim * 16);     // B*S*16 = 1024  (~2.4 MB total)

  pool_kernel  <<<Bdim * Tdim, 256, 0, stream>>>(x_enc, xbar);
  zseg_kernel  <<<Bdim * Sdim, 256, 0, stream>>>(xbar, zbar);
  scal_kernel  <<<1, 32, 0, stream>>>(enc_w, enc_b, gl_w1, gl_b1, gl_w2, gl_b2, gate, scal);
  whprep_kernel<<<(Tdim * PREDdim + 255) / 256, 256, 0, stream>>>(enc_w, head_w, Wh);
  cprep_kernel <<<1, 128, 0, stream>>>(enc_b, head_w, head_b, gate, Cp);
  topk_kernel  <<<Bdim * Sdim, 256, 0, stream>>>(zbar, topv, topi);
  mix_kernel   <<<Bdim * Sdim, 256, 0, stream>>>(xbar, zbar, topv, topi, scal, U);

  int tiles = ((Bdim * Ndim) / 16) * (PREDdim / 16); // 512 * 6 = 3072 waves
  head_kernel  <<<tiles * 32 / 256, 256, 0, stream>>>(U, Wh, Cp, out);
}